// Table_MHSA_RoPE_9680856285843
// MI455X (gfx1250) — compile-verified
//
#include <hip/hip_runtime.h>
#include <hip/hip_bf16.h>

// ---------------------------------------------------------------------------
// MHSA + RoPE for MI455X (gfx1250), fp32 via V_WMMA_F32_16X16X4_F32.
// B=2, N=2048, H=1024, heads=16, head_dim=64.
// Projection GEMMs use 32-row-per-wave tiles: two A-fragments per K-step with
// B-fragments reused across both row blocks (8 loads : 12 WMMAs in QKV).
// ---------------------------------------------------------------------------

typedef __attribute__((ext_vector_type(2))) float v2f;
typedef __attribute__((ext_vector_type(8))) float v8f;

#define BATCH 2
#define SEQ   2048
#define HID   1024
#define NHEAD 16
#define HDIM  64
// ln(10000)/32 : RoPE inverse-frequency exponent step
#define ROPE_LOG_STEP 0.28782313662425576f

static __device__ __forceinline__ v8f wmma_f32(v2f a, v2f b, v8f c) {
    return __builtin_amdgcn_wmma_f32_16x16x4_f32(
        /*neg_a=*/false, a, /*neg_b=*/false, b,
        /*c_mod=*/(short)0, c, /*reuse_a=*/false, /*reuse_b=*/false);
}

static __device__ __forceinline__ float half_max16(float v) {
    v = fmaxf(v, __shfl_xor(v, 1, 32));
    v = fmaxf(v, __shfl_xor(v, 2, 32));
    v = fmaxf(v, __shfl_xor(v, 4, 32));
    v = fmaxf(v, __shfl_xor(v, 8, 32));
    return v;
}
static __device__ __forceinline__ float half_sum16(float v) {
    v += __shfl_xor(v, 1, 32);
    v += __shfl_xor(v, 2, 32);
    v += __shfl_xor(v, 4, 32);
    v += __shfl_xor(v, 8, 32);
    return v;
}

// ---------------------------------------------------------------------------
// Kernel 1: fused Q/K/V projection + RoPE.
// Each wave: 32 rows x {cols [c0,c0+16) and [c0+32,c0+48)} of one head, for
// Q, K and V at once. Per K-step: 2 A-frag loads + 6 B-frag loads feed
// 12 WMMAs. The 32-apart column pairing makes rotate_half in-register.
// Outputs Q,K,V in [B, h, N, d] layout.
// ---------------------------------------------------------------------------
__global__ __launch_bounds__(128)
void qkv_rope_kernel(const float* __restrict__ x,
                     const float* __restrict__ wq,
                     const float* __restrict__ wk,
                     const float* __restrict__ wv,
                     float* __restrict__ Q,
                     float* __restrict__ K,
                     float* __restrict__ V) {
    const int wave  = (blockIdx.x * blockDim.x + threadIdx.x) >> 5; // 0..4095
    const int lane  = threadIdx.x & 31;
    const int mt    = wave >> 5;          // 128 row tiles (32 rows each)
    const int ct    = wave & 31;          // 32 column-strip pairs
    const int head  = ct >> 1;
    const int tbase = (ct & 1) * 16;      // 0 or 16 within the head's low half

    const int m  = lane & 15;             // A row within tile / B col within tile
    const int kh = lane >> 4;             // K-half (0: K0,K1; 1: K2,K3)

    const float* arow0 = x + (size_t)(mt * 32 + m) * HID + 2 * kh;
    const float* arow1 = arow0 + (size_t)16 * HID;
    const int    c0    = head * HDIM + tbase + m;  // low-half output feature
    const float* bq0   = wq + (size_t)c0 * HID + 2 * kh;
    const float* bq1   = wq + (size_t)(c0 + 32) * HID + 2 * kh;
    const float* bk0   = wk + (size_t)c0 * HID + 2 * kh;
    const float* bk1   = wk + (size_t)(c0 + 32) * HID + 2 * kh;
    const float* bv0   = wv + (size_t)c0 * HID + 2 * kh;
    const float* bv1   = wv + (size_t)(c0 + 32) * HID + 2 * kh;

    v8f aq0[2] = {}, aq1[2] = {}, ak0[2] = {}, ak1[2] = {}, av0[2] = {}, av1[2] = {};

    for (int k = 0; k < HID; k += 4) {
        v2f aA = *(const v2f*)(arow0 + k);
        v2f aB = *(const v2f*)(arow1 + k);
        v2f q0 = *(const v2f*)(bq0 + k);
        v2f q1 = *(const v2f*)(bq1 + k);
        v2f k0 = *(const v2f*)(bk0 + k);
        v2f k1 = *(const v2f*)(bk1 + k);
        v2f v0 = *(const v2f*)(bv0 + k);
        v2f v1 = *(const v2f*)(bv1 + k);
        aq0[0] = wmma_f32(aA, q0, aq0[0]);  aq0[1] = wmma_f32(aB, q0, aq0[1]);
        aq1[0] = wmma_f32(aA, q1, aq1[0]);  aq1[1] = wmma_f32(aB, q1, aq1[1]);
        ak0[0] = wmma_f32(aA, k0, ak0[0]);  ak0[1] = wmma_f32(aB, k0, ak0[1]);
        ak1[0] = wmma_f32(aA, k1, ak1[0]);  ak1[1] = wmma_f32(aB, k1, ak1[1]);
        av0[0] = wmma_f32(aA, v0, av0[0]);  av0[1] = wmma_f32(aB, v0, av0[1]);
        av1[0] = wmma_f32(aA, v1, av1[0]);  av1[1] = wmma_f32(aB, v1, av1[1]);
    }

    // RoPE epilogue. C-layout: vgpr r, this lane -> row rb*16 + r + 8*kh,
    // output feature j = tbase + (lane&15) (acc0) and j+32 (acc1).
    const int   j   = tbase + m;                         // in [0,32)
    const float inv = __expf(-(float)j * ROPE_LOG_STEP); // theta^(-j/32)

#pragma unroll
    for (int rb = 0; rb < 2; ++rb) {
#pragma unroll
        for (int r = 0; r < 8; ++r) {
            const int row = mt * 32 + rb * 16 + r + 8 * kh;
            const int pos = row & (SEQ - 1);
            const int b   = row >> 11;
            float s, c;
            __sincosf((float)pos * inv, &s, &c);

            const float qlo = aq0[rb][r], qhi = aq1[rb][r];
            const float klo = ak0[rb][r], khi = ak1[rb][r];
            const float qlo_r = qlo * c - qhi * s;
            const float qhi_r = qhi * c + qlo * s;
            const float klo_r = klo * c - khi * s;
            const float khi_r = khi * c + klo * s;

            const size_t base = (((size_t)b * NHEAD + head) * SEQ + pos) * HDIM;
            Q[base + j]      = qlo_r;
            Q[base + j + 32] = qhi_r;
            K[base + j]      = klo_r;
            K[base + j + 32] = khi_r;
            V[base + j]      = av0[rb][r];
            V[base + j + 32] = av1[rb][r];
        }
    }
}

// ---------------------------------------------------------------------------
// Kernel 2: flash attention. 4 waves/block; each wave owns 16 query rows and
// streams 16-key tiles with online softmax. K/V tiles staged in LDS; P is
// redistributed C-layout -> A-layout through a per-wave LDS scratch.
// Output O in [B, N, h, d] layout (row-major [B*N, 1024] for the final GEMM).
// ---------------------------------------------------------------------------
__global__ __launch_bounds__(128)
void flash_attn_kernel(const float* __restrict__ Q,
                       const float* __restrict__ K,
                       const float* __restrict__ V,
                       const int*  __restrict__ amask,
                       float* __restrict__ O) {
    __shared__ __align__(16) float Ks[16 * HDIM];
    __shared__ __align__(16) float Vs[16 * HDIM];
    __shared__ __align__(16) float Ps[4 * 16 * 16];

    const int bh    = blockIdx.x >> 5;       // 0..31  (b*16 + head)
    const int qblk  = blockIdx.x & 31;       // 0..31  (64 q-rows each)
    const int b     = bh >> 4;
    const int head  = bh & 15;
    const int waveI = threadIdx.x >> 5;
    const int lane  = threadIdx.x & 31;
    const int m     = lane & 15;
    const int kh    = lane >> 4;
    const int tid   = threadIdx.x;

    const float* Qbh = Q + (size_t)bh * SEQ * HDIM;
    const float* Kbh = K + (size_t)bh * SEQ * HDIM;
    const float* Vbh = V + (size_t)bh * SEQ * HDIM;
    const int*   mrow = amask + b * SEQ;
    const int    q0  = qblk * 64 + waveI * 16;

    // Q tile in A-fragment layout, pre-scaled by 1/sqrt(d).
    v2f qf[16];
#pragma unroll
    for (int t = 0; t < 16; ++t) {
        v2f q = *(const v2f*)(Qbh + (size_t)(q0 + m) * HDIM + 4 * t + 2 * kh);
        q.x *= 0.125f; q.y *= 0.125f;
        qf[t] = q;
    }

    v8f o0 = {}, o1 = {}, o2 = {}, o3 = {};
    float mi[8], li[8];
#pragma unroll
    for (int r = 0; r < 8; ++r) { mi[r] = -3.0e38f; li[r] = 0.0f; }

    float* pScr = Ps + waveI * 256;

    for (int kt = 0; kt < SEQ / 16; ++kt) {
        const int kbase = kt * 16;

        __syncthreads();                       // previous tile fully consumed
        {   // cooperative stage: 1024 floats each, 8 floats per thread
            const float4* ksrc = (const float4*)(Kbh + (size_t)kbase * HDIM);
            const float4* vsrc = (const float4*)(Vbh + (size_t)kbase * HDIM);
            float4* kdst = (float4*)Ks;
            float4* vdst = (float4*)Vs;
            kdst[tid * 2]     = ksrc[tid * 2];
            kdst[tid * 2 + 1] = ksrc[tid * 2 + 1];
            vdst[tid * 2]     = vsrc[tid * 2];
            vdst[tid * 2 + 1] = vsrc[tid * 2 + 1];
            if (kt + 1 < SEQ / 16)             // hint next tile into cache
                __builtin_prefetch(Kbh + (size_t)(kbase + 16) * HDIM, 0, 0);
        }
        __syncthreads();

        // S = Q * K^T  (16 WMMAs)
        v8f sA = {};
#pragma unroll
        for (int t = 0; t < 16; ++t) {
            v2f kf = *(const v2f*)&Ks[m * HDIM + 4 * t + 2 * kh];
            sA = wmma_f32(qf[t], kf, sA);
        }

        // column mask (this lane's key position)
        const int valid = mrow[kbase + m];

        // online softmax; row r lives in vgpr r across this wave-half's lanes
#pragma unroll
        for (int r = 0; r < 8; ++r) {
            float s = valid ? sA[r] : -3.0e38f;
            const float rmax = half_max16(s);
            const float mnew = fmaxf(mi[r], rmax);
            const float alpha = __expf(mi[r] - mnew);
            const float p = __expf(s - mnew);
            li[r] = li[r] * alpha + half_sum16(p);
            mi[r] = mnew;
            o0[r] *= alpha; o1[r] *= alpha; o2[r] *= alpha; o3[r] *= alpha;
            // park P into per-wave LDS scratch (C-layout -> [row][col])
            pScr[(r + 8 * kh) * 16 + m] = p;
        }

        // O += P * V  (16 WMMAs); A-frags re-read from LDS in A layout
#pragma unroll
        for (int t = 0; t < 4; ++t) {
            v2f pf = *(const v2f*)&pScr[m * 16 + 4 * t + 2 * kh];
            const int k0 = 4 * t + 2 * kh;
            v2f b0, b1, b2, b3;
            b0.x = Vs[k0 * HDIM + 0  + m]; b0.y = Vs[(k0 + 1) * HDIM + 0  + m];
            b1.x = Vs[k0 * HDIM + 16 + m]; b1.y = Vs[(k0 + 1) * HDIM + 16 + m];
            b2.x = Vs[k0 * HDIM + 32 + m]; b2.y = Vs[(k0 + 1) * HDIM + 32 + m];
            b3.x = Vs[k0 * HDIM + 48 + m]; b3.y = Vs[(k0 + 1) * HDIM + 48 + m];
            o0 = wmma_f32(pf, b0, o0);
            o1 = wmma_f32(pf, b1, o1);
            o2 = wmma_f32(pf, b2, o2);
            o3 = wmma_f32(pf, b3, o3);
        }
    }

    // normalize + write O as [B, N, h, d]
#pragma unroll
    for (int r = 0; r < 8; ++r) {
        const float rcp = 1.0f / li[r];
        const int npos = q0 + r + 8 * kh;
        const size_t base = (((size_t)b * SEQ + npos) * NHEAD + head) * HDIM + m;
        O[base + 0]  = o0[r] * rcp;
        O[base + 16] = o1[r] * rcp;
        O[base + 32] = o2[r] * rcp;
        O[base + 48] = o3[r] * rcp;
    }
}

// ---------------------------------------------------------------------------
// Kernel 3: output projection  out = O @ Wo^T.  Each wave: 32x64 tile
// (2 A-frags + 4 B-frags -> 8 WMMAs per K-step).
// ---------------------------------------------------------------------------
__global__ __launch_bounds__(128)
void out_proj_kernel(const float* __restrict__ Oin,
                     const float* __restrict__ wo,
                     float* __restrict__ out) {
    const int wave = (blockIdx.x * blockDim.x + threadIdx.x) >> 5; // 0..2047
    const int lane = threadIdx.x & 31;
    const int mt   = wave >> 4;           // 128 row tiles (32 rows each)
    const int n64  = (wave & 15) * 64;    // 16 column quads
    const int m    = lane & 15;
    const int kh   = lane >> 4;

    const float* arow0 = Oin + (size_t)(mt * 32 + m) * HID + 2 * kh;
    const float* arow1 = arow0 + (size_t)16 * HID;
    const float* b0 = wo + (size_t)(n64 + 0  + m) * HID + 2 * kh;
    const float* b1 = wo + (size_t)(n64 + 16 + m) * HID + 2 * kh;
    const float* b2 = wo + (size_t)(n64 + 32 + m) * HID + 2 * kh;
    const float* b3 = wo + (size_t)(n64 + 48 + m) * HID + 2 * kh;

    v8f c0[2] = {}, c1[2] = {}, c2[2] = {}, c3[2] = {};
    for (int k = 0; k < HID; k += 4) {
        v2f aA = *(const v2f*)(arow0 + k);
        v2f aB = *(const v2f*)(arow1 + k);
        v2f w0 = *(const v2f*)(b0 + k);
        v2f w1 = *(const v2f*)(b1 + k);
        v2f w2 = *(const v2f*)(b2 + k);
        v2f w3 = *(const v2f*)(b3 + k);
        c0[0] = wmma_f32(aA, w0, c0[0]);  c0[1] = wmma_f32(aB, w0, c0[1]);
        c1[0] = wmma_f32(aA, w1, c1[0]);  c1[1] = wmma_f32(aB, w1, c1[1]);
        c2[0] = wmma_f32(aA, w2, c2[0]);  c2[1] = wmma_f32(aB, w2, c2[1]);
        c3[0] = wmma_f32(aA, w3, c3[0]);  c3[1] = wmma_f32(aB, w3, c3[1]);
    }

#pragma unroll
    for (int rb = 0; rb < 2; ++rb) {
#pragma unroll
        for (int r = 0; r < 8; ++r) {
            const size_t row = (size_t)(mt * 32 + rb * 16 + r + 8 * kh);
            float* dst = out + row * HID + n64 + m;
            dst[0]  = c0[rb][r];
            dst[16] = c1[rb][r];
            dst[32] = c2[rb][r];
            dst[48] = c3[rb][r];
        }
    }
}

// ---------------------------------------------------------------------------
extern "C" void kernel_launch(void* const* d_in, const int* in_sizes, int n_in,
                              void* d_out, int out_size, void* d_ws, size_t ws_size,
                              hipStream_t stream) {
    const float* feature = (const float*)d_in[0];
    const int*   amask   = (const int*)d_in[1];
    const float* wq      = (const float*)d_in[2];
    const float* wk      = (const float*)d_in[3];
    const float* wv      = (const float*)d_in[4];
    const float* wo      = (const float*)d_in[5];
    float* outp = (float*)d_out;

    float* ws = (float*)d_ws;
    const size_t perBuf = (size_t)BATCH * NHEAD * SEQ * HDIM;  // 4,194,304 floats
    float* Q = ws;
    float* K = ws + perBuf;
    float* V = ws + 2 * perBuf;
    float* O = ws + 3 * perBuf;

    // 1) fused QKV + RoPE: 4096 waves -> 1024 blocks of 128
    qkv_rope_kernel<<<1024, 128, 0, stream>>>(feature, wq, wk, wv, Q, K, V);

    // 2) flash attention: (B*h)=32 combos x 32 q-blocks -> 1024 blocks of 128
    flash_attn_kernel<<<1024, 128, 0, stream>>>(Q, K, V, amask, O);

    // 3) output projection: 2048 waves -> 512 blocks of 128
    out_proj_kernel<<<512, 128, 0, stream>>>(O, wo, outp);
}